// IMUKalmanFilter_476
// MI455X (gfx1250) — compile-verified
//
#include <hip/hip_runtime.h>
#include <hip/hip_bf16.h>
#include <math.h>

#define B_  1024
#define T_  16
#define NI_ 11

typedef __attribute__((ext_vector_type(2))) float v2f;
typedef __attribute__((ext_vector_type(8))) float v8f;

// ---------- scalar 3x3 helpers (lane-0 code) ----------
__device__ inline void skew3(const float* v, float* K) {
    K[0] = 0.f;   K[1] = -v[2]; K[2] = v[1];
    K[3] = v[2];  K[4] = 0.f;   K[5] = -v[0];
    K[6] = -v[1]; K[7] = v[0];  K[8] = 0.f;
}
__device__ inline void transpose3(const float* A, float* T) {
    for (int i = 0; i < 3; ++i)
        for (int j = 0; j < 3; ++j) T[i*3+j] = A[j*3+i];
}
__device__ inline void mat3mul(const float* A, const float* Bm, float* C) {
    for (int i = 0; i < 3; ++i)
        for (int j = 0; j < 3; ++j) {
            float acc = 0.f;
            for (int k = 0; k < 3; ++k) acc += A[i*3+k] * Bm[k*3+j];
            C[i*3+j] = acc;
        }
}
__device__ inline void mat3vec(const float* A, const float* v, float* r) {
    for (int i = 0; i < 3; ++i)
        r[i] = A[i*3+0]*v[0] + A[i*3+1]*v[1] + A[i*3+2]*v[2];
}
__device__ inline void exp_so3(const float* phi, float* R) {
    float t2 = phi[0]*phi[0] + phi[1]*phi[1] + phi[2]*phi[2];
    bool small = t2 < 1e-8f;
    float t2s = small ? 1.f : t2;
    float t = sqrtf(t2s);
    float A  = small ? (1.f - t2/6.f)   : (sinf(t)/t);
    float Bc = small ? (0.5f - t2/24.f) : ((1.f - cosf(t))/t2s);
    float K[9], KK[9];
    skew3(phi, K);
    mat3mul(K, K, KK);
    for (int i = 0; i < 9; ++i)
        R[i] = ((i % 4 == 0) ? 1.f : 0.f) + A*K[i] + Bc*KK[i];
}
__device__ inline void log_so3(const float* C, float* phi) {
    float tr = C[0] + C[4] + C[8];
    float c = (tr - 1.f)*0.5f;
    c = fminf(fmaxf(c, -1.f + 1e-7f), 1.f - 1e-7f);
    float t = acosf(c);
    bool small = t < 1e-4f;
    float s = small ? 1.f : sinf(t);
    float coef = small ? (0.5f + t*t/12.f) : (t/(2.f*s));
    phi[0] = coef*(C[7] - C[5]);
    phi[1] = coef*(C[2] - C[6]);
    phi[2] = coef*(C[3] - C[1]);
}
__device__ inline void jlinv(const float* phi, float* J) {
    float t2 = phi[0]*phi[0] + phi[1]*phi[1] + phi[2]*phi[2];
    bool small = t2 < 1e-8f;
    float t2s = small ? 1.f : t2;
    float t = sqrtf(t2s);
    float coef = small ? (1.f/12.f)
                       : (1.f/t2s - (1.f + cosf(t))/(2.f*t*sinf(t)));
    float K[9], KK[9];
    skew3(phi, K);
    mat3mul(K, K, KK);
    for (int i = 0; i < 9; ++i)
        J[i] = ((i % 4 == 0) ? 1.f : 0.f) - 0.5f*K[i] + coef*KK[i];
}
__device__ inline void setBlock(float* M, int r, int c, const float* B3, float s) {
    for (int i = 0; i < 3; ++i)
        for (int j = 0; j < 3; ++j) M[(r+i)*32 + c + j] = s*B3[i*3+j];
}
__device__ inline void setDiag3(float* M, int r, int c, float s) {
    for (int i = 0; i < 3; ++i) M[(r+i)*32 + c + i] = s;
}

// ---------- 32x32x32 fp32 GEMM on the WMMA pipe ----------
// D = A @ B      (transB=false)   or   D = A @ B^T   (transB=true)
// A, B, D: row-major 32x32 in LDS. Pads must be zero in the operands.
__device__ inline void mm32(const float* A, const float* Bm, float* D,
                            int lane, bool transB) {
    const int half = lane >> 4;   // 0 or 1
    const int l16  = lane & 15;
    for (int ti = 0; ti < 2; ++ti) {
        for (int tj = 0; tj < 2; ++tj) {
            v8f acc = {0.f, 0.f, 0.f, 0.f, 0.f, 0.f, 0.f, 0.f};
#pragma unroll
            for (int kc = 0; kc < 8; ++kc) {
                v2f a, b;
#pragma unroll
                for (int j = 0; j < 2; ++j) {
                    const int k = kc*4 + j + 2*half;   // ISA A/B fragment layout
                    a[j] = A[(ti*16 + l16)*32 + k];
                    b[j] = transB ? Bm[(tj*16 + l16)*32 + k]
                                  : Bm[k*32 + tj*16 + l16];
                }
                acc = __builtin_amdgcn_wmma_f32_16x16x4_f32(
                        false, a, false, b, (short)0, acc, false, false);
            }
#pragma unroll
            for (int v = 0; v < 8; ++v)
                D[(ti*16 + v + 8*half)*32 + tj*16 + l16] = acc[v];
        }
    }
}

// ---------- the EKF kernel: one wave per batch element ----------
__global__ __launch_bounds__(32)
void ekf_kernel(const float* __restrict__ imu,
                const float* __restrict__ Rimu,
                const float* __restrict__ prev_pose,
                const float* __restrict__ prev_state,
                const float* __restrict__ prev_covar,
                const float* __restrict__ vis,
                const float* __restrict__ visR,
                float* __restrict__ out_pose,
                float* __restrict__ out_state,
                float* __restrict__ out_cov) {
    __shared__ float sF[1024];    // F / H / U scratch
    __shared__ float sG[1024];    // G / K scratch
    __shared__ float sR[1024];    // Rimu padded (constant)
    __shared__ float sPhi[1024];
    __shared__ float sCov[1024];
    __shared__ float sT1[1024];
    __shared__ float sT2[1024];
    __shared__ float sc[96];
    // sc map: 0..2 g | 3..11 C | 12..14 r | 15..17 v | 18..20 bw | 21..23 ba
    //         24 lambda | 25 t_acc | 26..34 C_acc | 35..37 r_acc | 38..40 v_acc
    //         41..56 pose | 57..62 residual | 64..72 E or J | 88 dt

    const int b = blockIdx.x;
    const int lane = threadIdx.x;

    for (int i = lane; i < 1024; i += 32) { sCov[i] = 0.f; sR[i] = 0.f; }
    __syncthreads();
    for (int i = lane; i < 361; i += 32)
        sCov[(i/19)*32 + (i%19)] = prev_covar[(size_t)b*361 + i];
    for (int i = lane; i < 144; i += 32)
        sR[(i/12)*32 + (i%12)] = Rimu[i];
    if (lane < 25) sc[lane] = prev_state[(size_t)b*25 + lane];
    if (lane < 16) sc[41 + lane] = prev_pose[(size_t)b*16 + lane];
    __syncthreads();

    // outputs at timestep index 0 (initial pose/state/covar)
    if (lane < 16) out_pose[((size_t)b*17 + 0)*16 + lane] = prev_pose[(size_t)b*16 + lane];
    if (lane < 25) out_state[((size_t)b*17 + 0)*25 + lane] = prev_state[(size_t)b*25 + lane];
    for (int i = lane; i < 361; i += 32)
        out_cov[((size_t)b*17 + 0)*361 + i] = prev_covar[(size_t)b*361 + i];

    for (int k = 0; k < T_; ++k) {
        // ================= predict =================
        // u-mask: zero rows/cols 3..8 of covariance
        for (int idx = lane; idx < 1024; idx += 32) {
            int i = idx >> 5, j = idx & 31;
            if ((i >= 3 && i < 9) || (j >= 3 && j < 9)) sCov[idx] = 0.f;
        }
        if (lane == 0) {
            sc[25] = 0.f;                                        // t_acc
            for (int i = 0; i < 9; ++i) sc[26+i] = (i%4==0)?1.f:0.f; // C_acc = I
            for (int i = 0; i < 3; ++i) { sc[35+i] = 0.f; sc[38+i] = 0.f; }
        }
        __syncthreads();

        const float* imub = imu + ((size_t)b*T_ + k)*NI_*7;

        for (int s = 0; s < NI_ - 1; ++s) {
            for (int idx = lane; idx < 1024; idx += 32) { sF[idx] = 0.f; sG[idx] = 0.f; }
            __syncthreads();
            if (lane == 0) {
                float dt = imub[(s+1)*7 + 0] - imub[s*7 + 0];
                float w[3], aa[3];
                for (int i = 0; i < 3; ++i) {
                    w[i]  = imub[s*7 + 1 + i] - sc[18+i];
                    aa[i] = imub[s*7 + 4 + i] - sc[21+i];
                }
                float Ca[9];  for (int i = 0; i < 9; ++i) Ca[i] = sc[26+i];
                float CaT[9]; transpose3(Ca, CaT);
                float t_acc = sc[25];
                float tv[3], vv[3];
                for (int i = 0; i < 3; ++i) tv[i] = sc[15+i] - sc[i]*t_acc + sc[38+i];
                mat3vec(CaT, tv, vv);
                float wsk[9], vsk[9];
                skew3(w, wsk); skew3(vv, vsk);
                float dtw[3] = { dt*w[0], dt*w[1], dt*w[2] };
                float E[9]; exp_so3(dtw, E);
                // ---- F blocks ----
                setBlock(sF, 3, 3, wsk, -1.f);
                setDiag3(sF, 3, 12, -1.f);
                float Cv[9]; mat3mul(Ca, vsk, Cv);
                setBlock(sF, 6, 3, Cv, -1.f);
                setBlock(sF, 6, 9, Ca, 1.f);
                setBlock(sF, 9, 0, CaT, -1.f);
                float Cg[3]; mat3vec(CaT, &sc[0], Cg);
                float Cgsk[9]; skew3(Cg, Cgsk);
                setBlock(sF, 9, 3, Cgsk, -1.f);
                setBlock(sF, 9, 9, wsk, -1.f);
                setBlock(sF, 9, 12, vsk, -1.f);
                setDiag3(sF, 9, 15, -1.f);
                // ---- G blocks ----
                setDiag3(sG, 3, 0, -1.f);
                setBlock(sG, 9, 0, vsk, -1.f);
                setDiag3(sG, 9, 6, -1.f);
                setDiag3(sG, 12, 3, 1.f);
                setDiag3(sG, 15, 9, 1.f);
                // stash dt, E for all lanes
                sc[88] = dt;
                for (int i = 0; i < 9; ++i) sc[64+i] = E[i];
                // ---- accumulator updates (do not feed this step's matmuls) ----
                float dt2 = dt*dt;
                float t3[3], Caa[3];
                for (int i = 0; i < 3; ++i) t3[i] = dt2*aa[i];
                mat3vec(Ca, t3, Caa);
                for (int i = 0; i < 3; ++i) sc[35+i] += sc[38+i]*dt + 0.5f*Caa[i];
                for (int i = 0; i < 3; ++i) t3[i] = dt*aa[i];
                mat3vec(Ca, t3, Caa);
                for (int i = 0; i < 3; ++i) sc[38+i] += Caa[i];
                float Cn[9]; mat3mul(Ca, E, Cn);
                for (int i = 0; i < 9; ++i) sc[26+i] = Cn[i];
                sc[25] = t_acc + dt;
            }
            __syncthreads();
            mm32(sF, sF, sT1, lane, false);            // F @ F
            __syncthreads();
            {
                float dt = sc[88]; float h2 = 0.5f*dt*dt;
                for (int idx = lane; idx < 1024; idx += 32) {
                    int i = idx >> 5, j = idx & 31;
                    float e = (i == j && i < 19) ? 1.f : 0.f;
                    sPhi[idx] = e + dt*sF[idx] + h2*sT1[idx];
                }
            }
            __syncthreads();
            if (lane == 0) {
                for (int i = 6; i < 9; ++i)
                    for (int j = 12; j < 15; ++j) sPhi[i*32 + j] = 0.f;
                for (int i = 0; i < 3; ++i)
                    for (int j = 0; j < 3; ++j) {
                        float et = sc[64 + j*3 + i];   // E^T
                        sPhi[(3+i)*32 + 3 + j] = et;
                        sPhi[(9+i)*32 + 9 + j] = et;
                    }
            }
            __syncthreads();
            mm32(sPhi, sG, sT1, lane, false);          // Phi @ G
            __syncthreads();
            mm32(sT1, sR, sT2, lane, false);           // (PhiG) @ Rimu
            __syncthreads();
            mm32(sT2, sT1, sF, lane, true);            // Q/dt = (PhiG R)(PhiG)^T
            __syncthreads();
            mm32(sPhi, sCov, sT2, lane, false);        // Phi @ P
            __syncthreads();
            mm32(sT2, sPhi, sT1, lane, true);          // (Phi P) Phi^T
            __syncthreads();
            {
                float dt = sc[88];
                for (int idx = lane; idx < 1024; idx += 32) {
                    int i = idx >> 5, j = idx & 31;
                    float val = 0.f;
                    if (i < 19 && j < 19) {
                        int a_ = (i <= j) ? idx : (j*32 + i);   // force_sym (upper tri)
                        val = sT1[a_] + dt*sF[a_];
                    }
                    sCov[idx] = val;
                }
            }
            __syncthreads();
        }

        // ---- predict finalize ----
        if (lane == 0) {
            float t_acc = sc[25];
            float Ca[9];  for (int i = 0; i < 9; ++i) Ca[i] = sc[26+i];
            float CaT[9]; transpose3(Ca, CaT);
            float rp[3], tv[3], vp[3];
            for (int i = 0; i < 3; ++i)
                rp[i] = sc[15+i]*t_acc - 0.5f*sc[i]*t_acc*t_acc + sc[35+i];
            for (int i = 0; i < 3; ++i)
                tv[i] = sc[15+i] - sc[i]*t_acc + sc[38+i];
            mat3vec(CaT, tv, vp);
            for (int i = 0; i < 9; ++i) sc[3+i]  = Ca[i];   // C_p
            for (int i = 0; i < 3; ++i) { sc[12+i] = rp[i]; sc[15+i] = vp[i]; }
            // residual + inverse-Jacobian for H
            float phi3[3]; log_so3(Ca, phi3);
            const float* vb = vis + ((size_t)b*T_ + k)*6;
            for (int i = 0; i < 3; ++i) sc[57+i] = vb[i] - phi3[i];
            float l = sc[24];
            for (int i = 0; i < 3; ++i) sc[60+i] = vb[3+i] - l*sc[12+i];
            float nphi[3] = { -phi3[0], -phi3[1], -phi3[2] };
            float J[9]; jlinv(nphi, J);
            for (int i = 0; i < 9; ++i) sc[64+i] = J[i];
        }
        __syncthreads();

        // ================= update =================
        for (int idx = lane; idx < 1024; idx += 32) sF[idx] = 0.f;   // H
        __syncthreads();
        if (lane == 0) {
            for (int i = 0; i < 3; ++i)
                for (int j = 0; j < 3; ++j) sF[i*32 + 3 + j] = sc[64 + i*3 + j];
            float l = sc[24];
            for (int i = 0; i < 3; ++i) sF[(3+i)*32 + 6 + i] = l;
            for (int i = 0; i < 3; ++i) sF[(3+i)*32 + 18]    = sc[12+i];
        }
        __syncthreads();
        mm32(sF, sCov, sT1, lane, false);    // H @ P
        __syncthreads();
        mm32(sT1, sF, sT2, lane, true);      // S0 = (H P) H^T
        __syncthreads();
        if (lane == 0) {
            const float* Rv = visR + ((size_t)b*T_ + k)*36;
            float M[6][12];
            for (int i = 0; i < 6; ++i)
                for (int j = 0; j < 6; ++j) {
                    M[i][j]     = sT2[i*32 + j] + Rv[i*6 + j];
                    M[i][6 + j] = (i == j) ? 1.f : 0.f;
                }
            for (int c = 0; c < 6; ++c) {          // Gauss-Jordan, partial pivot
                int p = c; float bst = fabsf(M[c][c]);
                for (int r2 = c+1; r2 < 6; ++r2) {
                    float a_ = fabsf(M[r2][c]);
                    if (a_ > bst) { bst = a_; p = r2; }
                }
                if (p != c)
                    for (int j = 0; j < 12; ++j) { float t = M[c][j]; M[c][j] = M[p][j]; M[p][j] = t; }
                float inv = 1.f / M[c][c];
                for (int j = 0; j < 12; ++j) M[c][j] *= inv;
                for (int r2 = 0; r2 < 6; ++r2)
                    if (r2 != c) {
                        float f = M[r2][c];
                        for (int j = 0; j < 12; ++j) M[r2][j] -= f*M[c][j];
                    }
            }
            for (int i = 0; i < 6; ++i)
                for (int j = 0; j < 6; ++j) sT2[i*32 + j] = M[i][6 + j];   // S^-1
        }
        __syncthreads();
        mm32(sCov, sF, sT1, lane, true);     // P H^T
        __syncthreads();
        mm32(sT1, sT2, sG, lane, false);     // K = P H^T S^-1
        __syncthreads();
        if (lane == 0) {
            float e[19];
            for (int i = 0; i < 19; ++i) {
                float acc = 0.f;
                for (int j = 0; j < 6; ++j) acc += sG[i*32 + j]*sc[57+j];
                e[i] = acc;
            }
            for (int i = 0; i < 3; ++i) sc[i] += e[i];
            float dC[9]; exp_so3(&e[3], dC);
            float Cc[9]; for (int i = 0; i < 9; ++i) Cc[i] = sc[3+i];
            float Cn[9]; mat3mul(Cc, dC, Cn);
            for (int i = 0; i < 9; ++i) sc[3+i] = Cn[i];
            for (int i = 0; i < 3; ++i) sc[12+i] += e[6+i];
            for (int i = 0; i < 3; ++i) sc[15+i] += e[9+i];
            for (int i = 0; i < 3; ++i) sc[18+i] += e[12+i];
            for (int i = 0; i < 3; ++i) sc[21+i] += e[15+i];
            sc[24] += e[18];
        }
        __syncthreads();
        mm32(sG, sF, sT2, lane, false);      // K @ H
        __syncthreads();
        for (int idx = lane; idx < 1024; idx += 32) {
            int i = idx >> 5, j = idx & 31;
            float val = 0.f;
            if (i < 19 && j < 19) val = ((i == j) ? 1.f : 0.f) - sT2[idx];
            sT1[idx] = val;                  // I - K H
        }
        __syncthreads();
        mm32(sT1, sCov, sT2, lane, false);   // est_P = (I-KH) P   -> sT2
        __syncthreads();

        // ================= composition =================
        for (int idx = lane; idx < 1024; idx += 32) sF[idx] = 0.f;   // U
        __syncthreads();
        if (lane == 0) {
            float Cc[9]; for (int i = 0; i < 9; ++i) Cc[i] = sc[3+i];
            float CT[9]; transpose3(Cc, CT);
            float Rp[9], tp[3];
            for (int i = 0; i < 3; ++i)
                for (int j = 0; j < 3; ++j) Rp[i*3+j] = sc[41 + i*4 + j];
            for (int i = 0; i < 3; ++i) tp[i] = sc[41 + i*4 + 3];
            float Rn[9]; mat3mul(CT, Rp, Rn);
            float dtp[3]; for (int i = 0; i < 3; ++i) dtp[i] = tp[i] - sc[12+i];
            float tn[3]; mat3vec(CT, dtp, tn);
            for (int i = 0; i < 3; ++i)
                for (int j = 0; j < 3; ++j) sc[41 + i*4 + j] = Rn[i*3+j];
            for (int i = 0; i < 3; ++i) sc[41 + i*4 + 3] = tn[i];
            sc[41+12] = 0.f; sc[41+13] = 0.f; sc[41+14] = 0.f; sc[41+15] = 1.f;
            float ng[3]; mat3vec(CT, &sc[0], ng);
            for (int i = 0; i < 3; ++i) sc[i] = ng[i];
            // U blocks
            for (int i = 0; i < 3; ++i)
                for (int j = 0; j < 3; ++j) sF[i*32 + j] = CT[i*3+j];
            float ngsk[9]; skew3(ng, ngsk);
            for (int i = 0; i < 3; ++i)
                for (int j = 0; j < 3; ++j) sF[i*32 + 3 + j] = ngsk[i*3+j];
            for (int i = 3; i < 19; ++i) sF[i*32 + i] = 1.f;
        }
        __syncthreads();
        mm32(sF, sT2, sT1, lane, false);     // U @ estP
        __syncthreads();
        mm32(sT1, sF, sT2, lane, true);      // (U estP) U^T
        __syncthreads();
        for (int idx = lane; idx < 1024; idx += 32) {
            int i = idx >> 5, j = idx & 31;
            float val = 0.f;
            if (i < 19 && j < 19) val = (i <= j) ? sT2[idx] : sT2[j*32 + i];
            sCov[idx] = val;                 // force_sym
        }
        __syncthreads();

        // ================= outputs at k+1 =================
        if (lane < 16) out_pose[((size_t)b*17 + k + 1)*16 + lane]  = sc[41 + lane];
        if (lane < 25) out_state[((size_t)b*17 + k + 1)*25 + lane] = sc[lane];
        for (int i = lane; i < 361; i += 32)
            out_cov[((size_t)b*17 + k + 1)*361 + i] = sCov[(i/19)*32 + (i%19)];

        if (k + 1 < T_) {
            const float* nxt = imu + ((size_t)b*T_ + (k + 1))*NI_*7;
            __builtin_prefetch(nxt, 0, 1);   // global_prefetch_b8
        }
        __syncthreads();
    }
}

extern "C" void kernel_launch(void* const* d_in, const int* in_sizes, int n_in,
                              void* d_out, int out_size, void* d_ws, size_t ws_size,
                              hipStream_t stream) {
    const float* imu        = (const float*)d_in[0];  // [B,T,NI,7]
    const float* Rimu       = (const float*)d_in[1];  // [12,12]
    const float* prev_pose  = (const float*)d_in[2];  // [B,4,4]
    const float* prev_state = (const float*)d_in[3];  // [B,25]
    const float* prev_covar = (const float*)d_in[4];  // [B,19,19]
    const float* vis        = (const float*)d_in[5];  // [B,T,6,1]
    const float* visR       = (const float*)d_in[6];  // [B,T,6,6]
    // d_in[7] = T_imu_cam: unused by the reference

    float* out_pose  = (float*)d_out;                           // [B,17,4,4]
    float* out_state = out_pose  + (size_t)B_*17*16;            // [B,17,25]
    float* out_cov   = out_state + (size_t)B_*17*25;            // [B,17,19,19]

    ekf_kernel<<<B_, 32, 0, stream>>>(imu, Rimu, prev_pose, prev_state,
                                      prev_covar, vis, visR,
                                      out_pose, out_state, out_cov);
}